// CA_event_2465311228659
// MI455X (gfx1250) — compile-verified
//
#include <hip/hip_runtime.h>

// ---- problem constants (match reference) ----
#define DIM           100
#define STRIDE        400          // 4*DIM: row stride of state and out
#define ROWS_PER_WAVE 16
#define WAVES_PER_BLK 8
#define ROWS_PER_BLK  (ROWS_PER_WAVE * WAVES_PER_BLK)   // 128
#define NTILES        7            // ceil(DIM/16): N tiles (cols 100..111 of tile 6 discarded)
#define KSTEPS        25           // K=100 consumed 4/step by wmma_f32_16x16x4 -> exactly 25

// LDS strides chosen for bank behavior (64 banks x 4B) + alignment:
//  A_STR/S_STR = 108: ml*108 mod 64 = ml*44 -> 16 distinct banks (gcd(44,64)=4),
//    all =0 mod 4, so the +2-DW offset of lanes 16..31 never collides: conflict-free
//    b64 fragment loads. 108*4 = 432 B keeps rows 16B-aligned for b128 stores.
//  B_STR = 116: 8*116 mod 64 = 32 -> the two half-waves of C-init/D-store hit
//    disjoint bank halves: conflict-free b32 row accesses. 464 B rows, 16B-aligned.
#define A_STR 108
#define S_STR 108
#define B_STR 116

typedef __attribute__((ext_vector_type(2))) float v2f;
typedef __attribute__((ext_vector_type(4))) float v4f;
typedef __attribute__((ext_vector_type(8))) float v8f;

__global__ __launch_bounds__(256) void ca_fused_wmma_kernel(
    const float* __restrict__ state,
    const float* __restrict__ A,
    const float* __restrict__ target,
    float* __restrict__ out)
{
    // Total LDS: 43200 + 55296 + 59392 = 157888 B -> 2 blocks/WGP (320 KB) = 4 waves/SIMD
    __shared__ __attribute__((aligned(16))) float A_lds[DIM * A_STR];                          // A[n][k]
    __shared__ __attribute__((aligned(16))) float s_lds[WAVES_PER_BLK][ROWS_PER_WAVE][S_STR];  // s
    __shared__ __attribute__((aligned(16))) float b_lds[WAVES_PER_BLK][ROWS_PER_WAVE][B_STR];  // base / D

    const int tid  = threadIdx.x;
    const int wave = tid >> 5;
    const int lane = tid & 31;
    const int hi   = lane >> 4;    // half-wave: K offset for A/B frags, M offset for C/D
    const int ml   = lane & 15;    // M for A-frag; N-within-tile for B/C/D frags

    // ---- cooperative stage: A[n][k] -> LDS (B-operand source: D = s@A^T => B(k,n)=A[n][k]) ----
    for (int i = tid; i < DIM * DIM; i += 256) {
        const int n = i / DIM;
        const int k = i - n * DIM;
        A_lds[n * A_STR + k] = A[i];
    }
    __syncthreads();

    const long rowBase = (long)blockIdx.x * ROWS_PER_BLK + (long)wave * ROWS_PER_WAVE;
    const v4f  zero4   = (v4f){0.0f, 0.0f, 0.0f, 0.0f};
    const int  c0      = 4 * lane;            // each lane owns a 4-column chunk

    v4f tv = zero4;
    if (lane < 25) tv = *(const v4f*)&target[c0];

    // ---- phase 1 (per wave): s = x^2/(1+x^2); base = u*s - x; b128 in/out.
    //      Lanes 25..27 produce base = 0 -> zero pad for b_lds cols 100..111.
    //      Per-wave LDS: DScnt waits suffice, no barrier needed. ----
    for (int r = 0; r < ROWS_PER_WAVE; ++r) {
        const float* srow = state + (rowBase + r) * STRIDE;
        if (lane < 28) {
            v4f xv = zero4, exv = zero4, wav = zero4;
            if (lane < 25) {
                xv  = *(const v4f*)&srow[c0];
                exv = *(const v4f*)&srow[DIM + c0];
                wav = *(const v4f*)&srow[3 * DIM + c0];
            }
            v4f sv, bv;
#pragma unroll
            for (int j = 0; j < 4; ++j) {
                const float x  = xv[j];
                const float x2 = x * x;
                const float s  = x2 / (1.0f + x2);
                const float u  = wav[j] * (x + exv[j] - tv[j]);
                sv[j] = s;
                bv[j] = fmaf(u, s, -x);        // u*s - B_COEF*x, B_COEF = 1
            }
            if (lane < 25) *(v4f*)&s_lds[wave][r][c0] = sv;
            *(v4f*)&b_lds[wave][r][c0] = bv;
        }
    }

    // ---- phase 2: D = s @ A^T + base via V_WMMA_F32_16X16X4_F32 ----
    // C/D layout: VGPR j -> M = j + 8*hi, lane ml -> N = nt*16 + ml (ISA 7.12.2)
    v8f acc[NTILES];
#pragma unroll
    for (int nt = 0; nt < NTILES; ++nt) {
        v8f c;
#pragma unroll
        for (int j = 0; j < 8; ++j)
            c[j] = b_lds[wave][j + 8 * hi][nt * 16 + ml];   // cols >= 100 are zeros
        acc[nt] = c;
    }

    const float* arow = &s_lds[wave][ml][0];
    const float* brow[NTILES];
#pragma unroll
    for (int nt = 0; nt < NTILES; ++nt) {
        int n = nt * 16 + ml;
        if (n >= DIM) n = DIM - 1;             // clamp: garbage only feeds discarded D cols
        brow[nt] = &A_lds[n * A_STR];
    }

#pragma unroll
    for (int ks = 0; ks < KSTEPS; ++ks) {
        const int k0 = ks * 4 + 2 * hi;        // A-frag 16x4: lane ml = M, v2f = K pair
        const v2f a = *(const v2f*)(arow + k0);
#pragma unroll
        for (int nt = 0; nt < NTILES; ++nt) {
            const v2f b = *(const v2f*)(brow[nt] + k0);   // B-frag 4x16: lane ml = N
            acc[nt] = __builtin_amdgcn_wmma_f32_16x16x4_f32(
                /*neg_a=*/false, a, /*neg_b=*/false, b,
                /*c_mod=*/(short)0, acc[nt], /*reuse_a=*/false, /*reuse_b=*/false);
        }
    }

    // ---- phase 3: D tiles -> LDS (reuse b_lds), then coalesced b128 global writes ----
#pragma unroll
    for (int nt = 0; nt < NTILES; ++nt)
#pragma unroll
        for (int j = 0; j < 8; ++j)
            b_lds[wave][j + 8 * hi][nt * 16 + ml] = acc[nt][j];

    for (int r = 0; r < ROWS_PER_WAVE; ++r) {
        float* orow = out + (rowBase + r) * STRIDE;
        if (lane < 25) {
            const v4f v = *(const v4f*)&b_lds[wave][r][c0];
            *(v4f*)&orow[c0]       = v;                          // dx
            *(v4f*)&orow[DIM + c0] = (v4f){-v[0], -v[1], -v[2], -v[3]};  // -dx
        }
        for (int q = lane; q < (2 * DIM) / 4; q += 32)           // cols 200..399 = zeros
            *(v4f*)&orow[2 * DIM + 4 * q] = zero4;
    }
}

extern "C" void kernel_launch(void* const* d_in, const int* in_sizes, int n_in,
                              void* d_out, int out_size, void* d_ws, size_t ws_size,
                              hipStream_t stream)
{
    const float* state  = (const float*)d_in[0];
    const float* A      = (const float*)d_in[1];
    const float* target = (const float*)d_in[2];
    float* out = (float*)d_out;

    const int batch  = in_sizes[0] / STRIDE;        // 131072
    const int blocks = batch / ROWS_PER_BLK;        // 1024 (batch divisible by 128)

    ca_fused_wmma_kernel<<<blocks, 256, 0, stream>>>(state, A, target, out);
}